// Linear_49563922596038
// MI455X (gfx1250) — compile-verified
//
#include <hip/hip_runtime.h>

typedef __attribute__((ext_vector_type(2))) float v2f;
typedef __attribute__((ext_vector_type(8))) float v8f;

#define D_IN  16
#define D_OUT 16

// ---------------------------------------------------------------- utilities
__global__ void zero_bins_kernel(int* __restrict__ counts,
                                 int* __restrict__ cursors, int n) {
    int i = blockIdx.x * blockDim.x + threadIdx.x;
    if (i < n) { counts[i] = 0; cursors[i] = 0; }
}

__global__ void hist_kernel(const int* __restrict__ widx,
                            int* __restrict__ counts, int e) {
    int i = blockIdx.x * blockDim.x + threadIdx.x;
    if (i < e) atomicAdd(&counts[widx[i]], 1);
}

// Single-block chunked inclusive scan -> exclusive offsets[0..n]
__global__ void scan_kernel(const int* __restrict__ counts,
                            int* __restrict__ offsets, int n) {
    __shared__ int buf[256];
    __shared__ int carry;
    if (threadIdx.x == 0) { carry = 0; offsets[0] = 0; }
    __syncthreads();
    int nch = (n + 255) / 256;
    for (int c = 0; c < nch; ++c) {
        int i = c * 256 + (int)threadIdx.x;
        buf[threadIdx.x] = (i < n) ? counts[i] : 0;
        __syncthreads();
        for (int off = 1; off < 256; off <<= 1) {
            int t = (threadIdx.x >= (unsigned)off) ? buf[threadIdx.x - off] : 0;
            __syncthreads();
            buf[threadIdx.x] += t;
            __syncthreads();
        }
        if (i < n) offsets[i + 1] = carry + buf[threadIdx.x];
        __syncthreads();
        if (threadIdx.x == 0) carry += buf[255];
        __syncthreads();
    }
}

// Scatter fused pairs: (edge_id, value_row). Kills one dependent-gather level
// in the hot kernel.
__global__ void scatter_kernel(const int* __restrict__ widx,
                               const int* __restrict__ iidx,
                               const int* __restrict__ offsets,
                               int* __restrict__ cursors,
                               int2* __restrict__ pairs, int e) {
    int i = blockIdx.x * blockDim.x + threadIdx.x;
    if (i < e) {
        int w = widx[i];
        int pos = offsets[w] + atomicAdd(&cursors[w], 1);
        pairs[pos] = make_int2(i, iidx[i]);
    }
}

// ------------------------------------------------------- WMMA tile kernel
// One block per weight bin; 8 waves/block; each wave owns 16-edge tiles.
// Y[16e x 16o] = X[16e x 16i] * W^T[16i x 16o] via 4x V_WMMA_F32_16X16X4_F32.
__global__ void __launch_bounds__(256)
linear_wmma_bin_kernel(const float* __restrict__ values,
                       const float* __restrict__ weight,
                       const int2*  __restrict__ pairs,
                       const int*   __restrict__ offsets,
                       float* __restrict__ out, int n_w) {
    int bin = blockIdx.x;
    int beg = offsets[bin];
    int end = offsets[bin + 1];
    int cnt = end - beg;
    if (cnt == 0) return;

    const int lane = threadIdx.x & 31;
    const int wv   = threadIdx.x >> 5;
    const int nwv  = blockDim.x >> 5;
    const int n    = lane & 15;   // matrix column (N) / A-row (M)
    const int h    = lane >> 4;   // half-wave selector

    // B fragments = W^T, shared across the whole bin.
    // f32 operand layout: lane element (row/col = lane&15), VGPR j <-> K = j + 2*h.
    const float* wbase = weight + (size_t)bin * (D_OUT * D_IN); // [o][i] row-major
    v2f bfrag[4];
#pragma unroll
    for (int kc = 0; kc < 4; ++kc) {
        const float* p = wbase + n * D_IN + 4 * kc + 2 * h;  // B[k][n] = W[n][k]
        bfrag[kc].x = p[0];
        bfrag[kc].y = p[1];
    }

    const int nfull = cnt >> 4;

    // ---- full tiles: no predication anywhere, EXEC stays all-ones ----
    for (int t = wv; t < nfull; t += nwv) {
        int tbase = beg + (t << 4);
        int2 pr = pairs[tbase + n];                 // (edge_id, value_row)
        const float* xp = values + (size_t)pr.y * D_IN;

        v2f afrag[4];
#pragma unroll
        for (int kc = 0; kc < 4; ++kc) {
            const float* p = xp + 4 * kc + 2 * h;
            afrag[kc].x = p[0];
            afrag[kc].y = p[1];
        }

        v8f acc = {};
#pragma unroll
        for (int kc = 0; kc < 4; ++kc) {
            acc = __builtin_amdgcn_wmma_f32_16x16x4_f32(
                false, afrag[kc], false, bfrag[kc],
                (short)0, acc, false, false);
        }

        // D layout: VGPR j holds row m=j (lanes 0-15) / m=j+8 (lanes 16-31), col n.
        // Edge ids come from lane r via ds_bpermute (no memory reload).
#pragma unroll
        for (int j = 0; j < 8; ++j) {
            int r  = j + 8 * h;
            int se = __shfl(pr.x, r, 32);
            out[(size_t)se * D_OUT + n] = acc[j];
        }
    }

    // ---- tail tile (if any): one wave, predicated stores only ----
    int rem = cnt & 15;
    if (rem && (nfull % nwv) == wv) {
        int tbase = beg + (nfull << 4);
        int rowp  = tbase + n;
        if (n >= rem) rowp = tbase;                 // clamp (cndmask, no EXEC change)
        int2 pr = pairs[rowp];
        const float* xp = values + (size_t)pr.y * D_IN;

        v2f afrag[4];
#pragma unroll
        for (int kc = 0; kc < 4; ++kc) {
            const float* p = xp + 4 * kc + 2 * h;
            afrag[kc].x = p[0];
            afrag[kc].y = p[1];
        }

        v8f acc = {};
#pragma unroll
        for (int kc = 0; kc < 4; ++kc) {
            acc = __builtin_amdgcn_wmma_f32_16x16x4_f32(
                false, afrag[kc], false, bfrag[kc],
                (short)0, acc, false, false);
        }

#pragma unroll
        for (int j = 0; j < 8; ++j) {
            int r  = j + 8 * h;
            int se = __shfl(pr.x, r, 32);
            if (r < rem) out[(size_t)se * D_OUT + n] = acc[j];
        }
    }
}

// ------------------------------------------------ scalar fallback (tiny ws)
__global__ void linear_fallback_kernel(const float* __restrict__ values,
                                       const float* __restrict__ weight,
                                       const int* __restrict__ iidx,
                                       const int* __restrict__ widx,
                                       float* __restrict__ out, int e) {
    int i = blockIdx.x * blockDim.x + threadIdx.x;
    if (i >= e) return;
    const float* x = values + (size_t)iidx[i] * D_IN;
    const float* w = weight + (size_t)widx[i] * (D_OUT * D_IN);
    float xr[D_IN];
#pragma unroll
    for (int k = 0; k < D_IN; ++k) xr[k] = x[k];
    float* o = out + (size_t)i * D_OUT;
#pragma unroll
    for (int oo = 0; oo < D_OUT; ++oo) {
        float acc = 0.f;
#pragma unroll
        for (int k = 0; k < D_IN; ++k) acc = fmaf(w[oo * D_IN + k], xr[k], acc);
        o[oo] = acc;
    }
}

// ---------------------------------------------------------------- launcher
extern "C" void kernel_launch(void* const* d_in, const int* in_sizes, int n_in,
                              void* d_out, int out_size, void* d_ws, size_t ws_size,
                              hipStream_t stream) {
    const float* values = (const float*)d_in[0];
    const float* weight = (const float*)d_in[1];
    const int*   iidx   = (const int*)d_in[2];
    const int*   widx   = (const int*)d_in[3];
    float*       out    = (float*)d_out;

    const int E  = in_sizes[2];
    const int NW = in_sizes[1] / (D_OUT * D_IN);

    // ws layout: counts[NW] | cursors[NW] | offsets[NW+1] | pad | pairs[E] (int2)
    size_t hdrInts = (size_t)3 * NW + 1;
    hdrInts = (hdrInts + 1) & ~(size_t)1;           // 8B-align the pairs array
    size_t need = hdrInts * sizeof(int) + (size_t)E * sizeof(int2);
    if (NW <= 0 || ws_size < need) {
        linear_fallback_kernel<<<(E + 255) / 256, 256, 0, stream>>>(
            values, weight, iidx, widx, out, E);
        return;
    }

    int*  counts  = (int*)d_ws;
    int*  cursors = counts + NW;
    int*  offsets = cursors + NW;
    int2* pairs   = (int2*)((int*)d_ws + hdrInts);

    zero_bins_kernel<<<(NW + 255) / 256, 256, 0, stream>>>(counts, cursors, NW);
    hist_kernel<<<(E + 255) / 256, 256, 0, stream>>>(widx, counts, E);
    scan_kernel<<<1, 256, 0, stream>>>(counts, offsets, NW);
    scatter_kernel<<<(E + 255) / 256, 256, 0, stream>>>(widx, iidx, offsets,
                                                        cursors, pairs, E);
    linear_wmma_bin_kernel<<<NW, 256, 0, stream>>>(
        values, weight, pairs, offsets, out, NW);
}